// GSKDB_1219770712834
// MI455X (gfx1250) — compile-verified
//
#include <hip/hip_runtime.h>
#include <hip/hip_bf16.h>
#include <math.h>

// ---------------------------------------------------------------------------
// MI455X (gfx1250) implementation. f32 end-to-end using V_WMMA_F32_16X16X4_F32
// for every GEMM/Gram (exact-precision match to the f32 reference). wave32,
// 16 waves (512 threads) per GEMM block; 64x128 macro-tiles, each wave owns
// two 16x16 subtiles sharing one A fragment. Tile staging is double-buffered
// through GLOBAL_LOAD_ASYNC_TO_LDS_B128 (ASYNCcnt) when the builtin exists.
// ---------------------------------------------------------------------------

typedef float v2f __attribute__((ext_vector_type(2)));
typedef float v4f __attribute__((ext_vector_type(4)));
typedef float v8f __attribute__((ext_vector_type(8)));
typedef int   v4i __attribute__((ext_vector_type(4)));

#if __has_builtin(__builtin_amdgcn_global_load_async_to_lds_b128)
#define ASYNC_STAGE 1
#else
#define ASYNC_STAGE 0
#endif

namespace {
constexpr int    kSCH = 256, kTCH = 512, kHW = 4096, kB = 8;
constexpr int    kS0 = 2432, kStride = 152;
constexpr float  kTemp = 4.0f;
constexpr float  kEps  = 1e-5f;
constexpr int    kNpix = kB * kHW;              // 32768 pixels for BN stats
constexpr int    kSB   = 76;                    // 76 * 32 = 2432 rows per pass block
constexpr size_t kLrow = (size_t)kS0 * kStride; // 369664 softmax row length

// LDS pitches: rows 16B-aligned (pitch % 4 == 0) and bank-conflict-free.
constexpr int PA_KM = 68;   // [kk][cc] A panels, 64 cols + 4 pad
constexpr int PB_KM = 132;  // [kk][nn] B panels, 128 cols + 4 pad
constexpr int P_MK  = 36;   // [row][kk] k-major tiles, 32 + 4 pad

// workspace layout (float offsets)
constexpr size_t OFF_M  = 0;                                  // 256x256 second moment
constexpr size_t OFF_SV = OFF_M  + 256 * 256;                 // 256 channel sums
constexpr size_t OFF_AL = OFF_SV + 256;                       // 512 bn scale
constexpr size_t OFF_MU = OFF_AL + 512;                       // 512 bn mean
constexpr size_t OFF_SG = OFF_MU + 512;                       // 256x2432 gathered student
constexpr size_t OFF_X1 = OFF_SG + (size_t)256 * 2432;        // 512x2432 conv1 out / relu
constexpr size_t OFF_Z  = OFF_X1 + (size_t)512 * 2432;        // 512x2432 conv2 out / norm
constexpr size_t OFF_TN = OFF_Z  + (size_t)512 * 2432;        // 512x2432 teacher norm
constexpr size_t OFF_TT = OFF_TN + (size_t)512 * 2432;        // 2432x2432 teacher gram
constexpr size_t OFF_SS = OFF_TT + (size_t)kS0 * kS0;         // 2432x2432 student gram
constexpr size_t OFF_P1 = OFF_SS + (size_t)kS0 * kS0;         // 2*16*76 max partials
constexpr size_t OFF_MA = OFF_P1 + 2 * 16 * kSB;              // 16 row maxima (tt)
constexpr size_t OFF_MB = OFF_MA + 16;                        // 16 row maxima (ss)
constexpr size_t OFF_P2 = OFF_MB + 16;                        // 3*16*76 sum partials
constexpr size_t OFF_KL = OFF_P2 + 3 * 16 * kSB;              // 16 per-row KL
} // namespace

// ---------------- async 16B global->LDS copy (fallback: vector copy) --------
__device__ __forceinline__ void cp16(float* l, const float* g) {
#if ASYNC_STAGE
  __builtin_amdgcn_global_load_async_to_lds_b128(
      (__attribute__((address_space(1))) v4i*)g,
      (__attribute__((address_space(3))) v4i*)l, 0, 0);
#else
  *(v4f*)l = *(const v4f*)g;
#endif
}
__device__ __forceinline__ void stage_fence() {
#if ASYNC_STAGE
#if __has_builtin(__builtin_amdgcn_s_wait_asynccnt)
  __builtin_amdgcn_s_wait_asynccnt(0);
#else
  asm volatile("s_wait_asynccnt 0" ::: "memory");
#endif
#endif
}

// ---------------- block reductions (power-of-2 blockDim) --------------------
__device__ __forceinline__ float block_sum(float v, float* red) {
  int t = threadIdx.x;
  red[t] = v; __syncthreads();
  for (int s = blockDim.x >> 1; s > 0; s >>= 1) {
    if (t < s) red[t] += red[t + s];
    __syncthreads();
  }
  float r = red[0]; __syncthreads();
  return r;
}
__device__ __forceinline__ float block_max(float v, float* red) {
  int t = threadIdx.x;
  red[t] = v; __syncthreads();
  for (int s = blockDim.x >> 1; s > 0; s >>= 1) {
    if (t < s) red[t] = fmaxf(red[t], red[t + s]);
    __syncthreads();
  }
  float r = red[0]; __syncthreads();
  return r;
}

// -------- one K=32 chunk: one A fragment feeds two 16x16 WMMA subtiles ------
// A frag (ISA 7.12.2, 16x4 f32): lanes 0-15 hold K=0,1; lanes 16-31 hold K=2,3.
template <int APITCH, bool A_KMAJOR, int BPITCH, bool B_KMAJOR>
__device__ __forceinline__ void chunk_dual(const float* As, const float* Bs,
                                           int am0, int bn0, int m, int h,
                                           v8f& acc0, v8f& acc1) {
#pragma unroll
  for (int kk = 0; kk < 32; kk += 4) {
    v2f a, b0, b1;
    if (A_KMAJOR) {
      a.x = As[(kk + 2 * h    ) * APITCH + am0 + m];
      a.y = As[(kk + 2 * h + 1) * APITCH + am0 + m];
    } else {
      a.x = As[(am0 + m) * APITCH + kk + 2 * h];
      a.y = As[(am0 + m) * APITCH + kk + 2 * h + 1];
    }
    if (B_KMAJOR) {
      b0.x = Bs[(kk + 2 * h    ) * BPITCH + bn0 + m];
      b0.y = Bs[(kk + 2 * h + 1) * BPITCH + bn0 + m];
      b1.x = Bs[(kk + 2 * h    ) * BPITCH + bn0 + 64 + m];
      b1.y = Bs[(kk + 2 * h + 1) * BPITCH + bn0 + 64 + m];
    } else {
      b0.x = Bs[(bn0 + m) * BPITCH + kk + 2 * h];
      b0.y = Bs[(bn0 + m) * BPITCH + kk + 2 * h + 1];
      b1.x = Bs[(bn0 + 64 + m) * BPITCH + kk + 2 * h];
      b1.y = Bs[(bn0 + 64 + m) * BPITCH + kk + 2 * h + 1];
    }
    acc0 = __builtin_amdgcn_wmma_f32_16x16x4_f32(false, a, false, b0, (short)0,
                                                 acc0, false, false);
    acc1 = __builtin_amdgcn_wmma_f32_16x16x4_f32(false, a, false, b1, (short)0,
                                                 acc1, false, false);
  }
}

// ---------------- 1) per-channel sum of student feats -----------------------
__global__ void svec_kernel(const float* __restrict__ s, float* __restrict__ svec) {
  __shared__ float red[256];
  const int c = blockIdx.x;
  float acc = 0.f;
  for (int b = 0; b < kB; ++b) {
    const float* p = s + (size_t)b * kSCH * kHW + (size_t)c * kHW;
    for (int i = threadIdx.x; i < kHW; i += 256) acc += p[i];
  }
  float r = block_sum(acc, red);
  if (threadIdx.x == 0) svec[c] = r;
}

// ---------------- 2) M = S S^T over all 32768 pixels (WMMA) -----------------
// grid (8 tiles of 64x128, 8 batches); partial tiles atomically accumulated.
// Both operands are k(=pixel)-contiguous in memory -> k-major LDS tiles.
__global__ void moment_kernel(const float* __restrict__ s, float* __restrict__ Mm) {
  __shared__ alignas(16) float As[2][64 * P_MK];
  __shared__ alignas(16) float Bs[2][128 * P_MK];
  const int bi = (blockIdx.x >> 1) * 64, bj = (blockIdx.x & 1) * 128;
  const float* base = s + (size_t)blockIdx.y * kSCH * kHW;
  const int t = threadIdx.x, lane = t & 31, w = t >> 5;
  const int am0 = (w >> 2) * 16, bn0 = (w & 3) * 16;
  const int m = lane & 15, h = lane >> 4;

  auto stage = [&](int k0, float* dA, float* dB) {
    { int cc = t >> 3, kk4 = (t & 7) * 4;          // 512 A quads
      cp16(&dA[cc * P_MK + kk4], &base[(size_t)(bi + cc) * kHW + k0 + kk4]); }
#pragma unroll
    for (int i = 0; i < 2; ++i) {                   // 1024 B quads
      int q = t + i * 512;
      int cc = q >> 3, kk4 = (q & 7) * 4;
      cp16(&dB[cc * P_MK + kk4], &base[(size_t)(bj + cc) * kHW + k0 + kk4]);
    }
  };

  v8f acc0 = {0.f, 0.f, 0.f, 0.f, 0.f, 0.f, 0.f, 0.f};
  v8f acc1 = acc0;
  stage(0, As[0], Bs[0]);
  stage_fence(); __syncthreads();
  int cur = 0;
  for (int k0 = 0; k0 < kHW; k0 += 32) {
    int nxt = cur ^ 1;
    if (k0 + 32 < kHW) stage(k0 + 32, As[nxt], Bs[nxt]);
    chunk_dual<P_MK, false, P_MK, false>(As[cur], Bs[cur], am0, bn0, m, h, acc0, acc1);
    stage_fence(); __syncthreads();
    cur = nxt;
  }
#pragma unroll
  for (int v = 0; v < 8; ++v) {
    atomicAdd(&Mm[(size_t)(bi + am0 + v + 8 * h) * 256 + bj + bn0 + m], acc0[v]);
    atomicAdd(&Mm[(size_t)(bi + am0 + v + 8 * h) * 256 + bj + bn0 + 64 + m], acc1[v]);
  }
}

// ---------------- 3) BN params via moment trick -----------------------------
// mu_o = w1[o].svec/N ; E[x^2]_o = w1[o]^T (M/N) w1[o] ; var = E2 - mu^2
__global__ void bnparam_kernel(const float* __restrict__ Mm, const float* __restrict__ svec,
                               const float* __restrict__ w1, const float* __restrict__ gamma,
                               float* __restrict__ alpha, float* __restrict__ mu) {
  __shared__ float red[256];
  const int o = blockIdx.x, c = threadIdx.x;
  const float* wrow = w1 + (size_t)o * kSCH;
  const float wv = wrow[c];
  const float* mrow = Mm + (size_t)c * kSCH;
  float tacc = 0.f;
  for (int d = 0; d < kSCH; ++d) tacc += mrow[d] * wrow[d];
  float s1 = block_sum(wv * svec[c], red);   // sum x
  float s2 = block_sum(wv * tacc, red);      // sum x^2
  if (c == 0) {
    float mval = s1 / (float)kNpix;
    float e2   = s2 / (float)kNpix;
    float var  = e2 - mval * mval;
    alpha[o] = gamma[o] * rsqrtf(var + kEps);
    mu[o] = mval;
  }
}

// ---------------- 4) gather sampled student columns -------------------------
__global__ void gather_student(const float* __restrict__ s, const int* __restrict__ pix,
                               float* __restrict__ Sg) {
  int i = blockIdx.x * 256 + threadIdx.x;        // 256*2432 exact
  int c = i / kS0, ss = i - c * kS0;
  Sg[i] = s[(size_t)7 * kSCH * kHW + (size_t)c * kHW + pix[ss]];
}

// ---------------- 5) teacher gather + per-pixel L2 normalize ----------------
__global__ void teacher_norm(const float* __restrict__ tf, const int* __restrict__ pix,
                             float* __restrict__ Tn) {
  __shared__ float red[256];
  const int s = blockIdx.x, t = threadIdx.x;
  const float* base = tf + (size_t)7 * kTCH * kHW + pix[s];
  float v0 = base[(size_t)t * kHW];
  float v1 = base[(size_t)(t + 256) * kHW];
  float ss = block_sum(v0 * v0 + v1 * v1, red);
  float inv = 1.f / fmaxf(sqrtf(ss), 1e-12f);
  Tn[(size_t)t * kS0 + s]         = v0 * inv;
  Tn[(size_t)(t + 256) * kS0 + s] = v1 * inv;
}

// ------- 6) C[512 x 2432] = W[512 x K] * Bm[K x 2432] (WMMA, 64x128 tiles) --
__global__ void gemm_kernel(const float* __restrict__ W, const float* __restrict__ Bm,
                            float* __restrict__ C, int K) {
  __shared__ alignas(16) float As[2][64 * P_MK];
  __shared__ alignas(16) float Bs[2][32 * PB_KM];
  const int o0 = blockIdx.x * 64, p0 = blockIdx.y * 128;
  const int t = threadIdx.x, lane = t & 31, w = t >> 5;
  const int am0 = (w >> 2) * 16, bn0 = (w & 3) * 16;
  const int m = lane & 15, h = lane >> 4;

  auto stage = [&](int k0, float* dA, float* dB) {
    { int mm = t >> 3, kk4 = (t & 7) * 4;          // A: k-contiguous quads
      cp16(&dA[mm * P_MK + kk4], &W[(size_t)(o0 + mm) * K + k0 + kk4]); }
#pragma unroll
    for (int i = 0; i < 2; ++i) {                   // B: n-contiguous quads
      int q = t + i * 512;
      int kk = q >> 5, nn4 = (q & 31) * 4;
      cp16(&dB[kk * PB_KM + nn4], &Bm[(size_t)(k0 + kk) * kS0 + p0 + nn4]);
    }
  };

  v8f acc0 = {0.f, 0.f, 0.f, 0.f, 0.f, 0.f, 0.f, 0.f};
  v8f acc1 = acc0;
  stage(0, As[0], Bs[0]);
  stage_fence(); __syncthreads();
  int cur = 0;
  for (int k0 = 0; k0 < K; k0 += 32) {
    int nxt = cur ^ 1;
    if (k0 + 32 < K) stage(k0 + 32, As[nxt], Bs[nxt]);
    chunk_dual<P_MK, false, PB_KM, true>(As[cur], Bs[cur], am0, bn0, m, h, acc0, acc1);
    stage_fence(); __syncthreads();
    cur = nxt;
  }
#pragma unroll
  for (int v = 0; v < 8; ++v) {
    C[(size_t)(o0 + am0 + v + 8 * h) * kS0 + p0 + bn0 + m]      = acc0[v];
    C[(size_t)(o0 + am0 + v + 8 * h) * kS0 + p0 + bn0 + 64 + m] = acc1[v];
  }
}

// ---------------- 7) BN (train) + ReLU in place -----------------------------
__global__ void bnrelu_kernel(float* __restrict__ X, const float* __restrict__ alpha,
                              const float* __restrict__ mu, const float* __restrict__ beta) {
  int i = blockIdx.x * 256 + threadIdx.x;        // 512*2432 exact
  int o = i / kS0;
  float x = X[i];
  X[i] = fmaxf(0.f, alpha[o] * (x - mu[o]) + beta[o]);
}

// ---------------- 8) per-pixel L2 normalize columns of Z in place -----------
__global__ void colnorm_kernel(float* __restrict__ Z) {
  __shared__ float red[256];
  const int s = blockIdx.x, t = threadIdx.x;
  float v0 = Z[(size_t)t * kS0 + s];
  float v1 = Z[(size_t)(t + 256) * kS0 + s];
  float ss = block_sum(v0 * v0 + v1 * v1, red);
  float inv = 1.f / fmaxf(sqrtf(ss), 1e-12f);
  Z[(size_t)t * kS0 + s]         = v0 * inv;
  Z[(size_t)(t + 256) * kS0 + s] = v1 * inv;
}

// --------- 9) tt = Tn^T Tn, ss = Zn^T Zn (WMMA, K=512, 64x128 tiles) --------
__global__ void gram_kernel(const float* __restrict__ Tn, const float* __restrict__ Zn,
                            float* __restrict__ tt, float* __restrict__ ssO) {
  __shared__ alignas(16) float As[2][32 * PA_KM];
  __shared__ alignas(16) float Bs[2][32 * PB_KM];
  const float* F = blockIdx.z ? Zn : Tn;
  float* out     = blockIdx.z ? ssO : tt;
  const int i0 = blockIdx.x * 64, j0 = blockIdx.y * 128;
  const int t = threadIdx.x, lane = t & 31, w = t >> 5;
  const int am0 = (w >> 2) * 16, bn0 = (w & 3) * 16;
  const int m = lane & 15, h = lane >> 4;

  auto stage = [&](int k0, float* dA, float* dB) {
    { int kk = t >> 4, cc4 = (t & 15) * 4;         // A: n-contiguous quads
      cp16(&dA[kk * PA_KM + cc4], &F[(size_t)(k0 + kk) * kS0 + i0 + cc4]); }
#pragma unroll
    for (int i = 0; i < 2; ++i) {                   // B: n-contiguous quads
      int q = t + i * 512;
      int kk = q >> 5, nn4 = (q & 31) * 4;
      cp16(&dB[kk * PB_KM + nn4], &F[(size_t)(k0 + kk) * kS0 + j0 + nn4]);
    }
  };

  v8f acc0 = {0.f, 0.f, 0.f, 0.f, 0.f, 0.f, 0.f, 0.f};
  v8f acc1 = acc0;
  stage(0, As[0], Bs[0]);
  stage_fence(); __syncthreads();
  int cur = 0;
  for (int k0 = 0; k0 < kTCH; k0 += 32) {
    int nxt = cur ^ 1;
    if (k0 + 32 < kTCH) stage(k0 + 32, As[nxt], Bs[nxt]);
    chunk_dual<PA_KM, true, PB_KM, true>(As[cur], Bs[cur], am0, bn0, m, h, acc0, acc1);
    stage_fence(); __syncthreads();
    cur = nxt;
  }
#pragma unroll
  for (int v = 0; v < 8; ++v) {
    out[(size_t)(i0 + am0 + v + 8 * h) * kS0 + j0 + bn0 + m]      = acc0[v];
    out[(size_t)(i0 + am0 + v + 8 * h) * kS0 + j0 + bn0 + 64 + m] = acc1[v];
  }
}

// ---------------- 10) pass 1: row maxima of the shuffled gathers ------------
__global__ void pass1_kernel(const float* __restrict__ tt, const float* __restrict__ ssM,
                             const int* __restrict__ shuf, float* __restrict__ p1) {
  __shared__ float red[256];
  const int r = blockIdx.y, sb = blockIdx.x;
  const int s_lo = sb * 32;
  float ma = -INFINITY, mb = -INFINITY;
  for (int idx = threadIdx.x; idx < 32 * kStride; idx += 256) {
    int ls = idx / kStride, j = idx - ls * kStride;
    int s = s_lo + ls;
    const int* srow = shuf + (size_t)s * kS0 + r * kStride;
    __builtin_prefetch(srow + j + 64, 0, 1);   // global_prefetch_b8
    int col = srow[j];
    ma = fmaxf(ma, tt[(size_t)s * kS0 + col]);
    mb = fmaxf(mb, ssM[(size_t)s * kS0 + col]);
  }
  float rma = block_max(ma, red);
  float rmb = block_max(mb, red);
  if (threadIdx.x == 0) {
    p1[r * kSB + sb] = rma;
    p1[16 * kSB + r * kSB + sb] = rmb;
  }
}

__global__ void reduce_max_kernel(const float* __restrict__ p1,
                                  float* __restrict__ Ma, float* __restrict__ Mb) {
  __shared__ float red[128];
  const int r = blockIdx.x, t = threadIdx.x;
  float va = (t < kSB) ? p1[r * kSB + t] : -INFINITY;
  float vb = (t < kSB) ? p1[16 * kSB + r * kSB + t] : -INFINITY;
  float rma = block_max(va, red);
  float rmb = block_max(vb, red);
  if (t == 0) { Ma[r] = rma; Mb[r] = rmb; }
}

// ---------------- 11) pass 2: Za, Zb and cross term W -----------------------
__global__ void pass2_kernel(const float* __restrict__ tt, const float* __restrict__ ssM,
                             const int* __restrict__ shuf, const float* __restrict__ Ma,
                             const float* __restrict__ Mb, float* __restrict__ p2) {
  __shared__ float red[256];
  const int r = blockIdx.y, sb = blockIdx.x;
  const int s_lo = sb * 32;
  const float ma = Ma[r], mb = Mb[r];
  float za = 0.f, zb = 0.f, wacc = 0.f;
  for (int idx = threadIdx.x; idx < 32 * kStride; idx += 256) {
    int ls = idx / kStride, j = idx - ls * kStride;
    int s = s_lo + ls;
    const int* srow = shuf + (size_t)s * kS0 + r * kStride;
    __builtin_prefetch(srow + j + 64, 0, 1);
    int col = srow[j];
    float va = tt[(size_t)s * kS0 + col];
    float vb = ssM[(size_t)s * kS0 + col];
    float ea = __expf((va - ma) / kTemp);
    za += ea;
    zb += __expf((vb - mb) / kTemp);
    wacc += ea * (va - vb) / kTemp;
  }
  float rza = block_sum(za, red);
  float rzb = block_sum(zb, red);
  float rw  = block_sum(wacc, red);
  if (threadIdx.x == 0) {
    p2[0 * 16 * kSB + r * kSB + sb] = rza;
    p2[1 * 16 * kSB + r * kSB + sb] = rzb;
    p2[2 * 16 * kSB + r * kSB + sb] = rw;
  }
}

// kl_r = W/Za - (Ma-Mb)/T - ln Za + ln Zb, scaled by T^2 / L
__global__ void reduce_kl_kernel(const float* __restrict__ p2, const float* __restrict__ Ma,
                                 const float* __restrict__ Mb, float* __restrict__ kl) {
  __shared__ float red[128];
  const int r = blockIdx.x, t = threadIdx.x;
  float za = (t < kSB) ? p2[0 * 16 * kSB + r * kSB + t] : 0.f;
  float zb = (t < kSB) ? p2[1 * 16 * kSB + r * kSB + t] : 0.f;
  float wv = (t < kSB) ? p2[2 * 16 * kSB + r * kSB + t] : 0.f;
  float sza = block_sum(za, red);
  float szb = block_sum(zb, red);
  float sw  = block_sum(wv, red);
  if (t == 0) {
    float v = sw / sza - (Ma[r] - Mb[r]) / kTemp - logf(sza) + logf(szb);
    kl[r] = v * (kTemp * kTemp) / (float)kLrow;
  }
}

__global__ void final_kernel(const float* __restrict__ kl, float* __restrict__ out) {
  __shared__ float red[32];
  float v = (threadIdx.x < 16) ? kl[threadIdx.x] : 0.f;
  float s = block_sum(v, red);
  if (threadIdx.x == 0) out[0] = s / 16.f;
}

// ---------------------------------------------------------------------------
extern "C" void kernel_launch(void* const* d_in, const int* in_sizes, int n_in,
                              void* d_out, int out_size, void* d_ws, size_t ws_size,
                              hipStream_t stream) {
  (void)in_sizes; (void)n_in; (void)out_size; (void)ws_size;
  const float* s_feats = (const float*)d_in[1];
  const float* t_feats = (const float*)d_in[2];
  const float* w1      = (const float*)d_in[4];
  const float* gamma   = (const float*)d_in[5];
  const float* beta    = (const float*)d_in[6];
  const float* w2      = (const float*)d_in[7];
  const int*   pix     = (const int*)d_in[8];   // [19*128] flat = [2432]
  const int*   shuf    = (const int*)d_in[9];   // [2432*2432]
  float* ws  = (float*)d_ws;
  float* out = (float*)d_out;

  float* Mm = ws + OFF_M;  float* sv = ws + OFF_SV;
  float* al = ws + OFF_AL; float* mu = ws + OFF_MU;
  float* Sg = ws + OFF_SG; float* X1 = ws + OFF_X1;
  float* Z  = ws + OFF_Z;  float* Tn = ws + OFF_TN;
  float* tt = ws + OFF_TT; float* ss = ws + OFF_SS;
  float* p1 = ws + OFF_P1; float* Ma = ws + OFF_MA; float* Mb = ws + OFF_MB;
  float* p2 = ws + OFF_P2; float* kl = ws + OFF_KL;

  (void)hipMemsetAsync(Mm, 0, 256 * 256 * sizeof(float), stream); // capture-safe memset node

  svec_kernel   <<<256, 256, 0, stream>>>(s_feats, sv);
  moment_kernel <<<dim3(8, 8), 512, 0, stream>>>(s_feats, Mm);
  bnparam_kernel<<<512, 256, 0, stream>>>(Mm, sv, w1, gamma, al, mu);

  gather_student<<<2432, 256, 0, stream>>>(s_feats, pix, Sg);
  teacher_norm  <<<2432, 256, 0, stream>>>(t_feats, pix, Tn);

  gemm_kernel   <<<dim3(8, 19), 512, 0, stream>>>(w1, Sg, X1, kSCH);  // conv1 @ samples
  bnrelu_kernel <<<4864, 256, 0, stream>>>(X1, al, mu, beta);
  gemm_kernel   <<<dim3(8, 19), 512, 0, stream>>>(w2, X1, Z, kTCH);   // conv2 @ samples
  colnorm_kernel<<<2432, 256, 0, stream>>>(Z);

  gram_kernel   <<<dim3(38, 19, 2), 512, 0, stream>>>(Tn, Z, tt, ss);

  pass1_kernel     <<<dim3(kSB, 16), 256, 0, stream>>>(tt, ss, shuf, p1);
  reduce_max_kernel<<<16, 128, 0, stream>>>(p1, Ma, Mb);
  pass2_kernel     <<<dim3(kSB, 16), 256, 0, stream>>>(tt, ss, shuf, Ma, Mb, p2);
  reduce_kl_kernel <<<16, 128, 0, stream>>>(p2, Ma, Mb, kl);
  final_kernel     <<<1, 32, 0, stream>>>(kl, out);
}